// MultiAnchor_63728724738221
// MI455X (gfx1250) — compile-verified
//
#include <hip/hip_runtime.h>
#include <hip/hip_bf16.h>

#define A_TOTAL 34000
#define NBOX    128
#define BLK     256

typedef unsigned int tdm_u4 __attribute__((ext_vector_type(4)));
typedef int          tdm_i8 __attribute__((ext_vector_type(8)));
typedef int          tdm_i4 __attribute__((ext_vector_type(4)));

__global__ __launch_bounds__(BLK)
void MultiAnchor_encode_kernel(const float* __restrict__ boxes,  // [B, N, 4] yxyx
                               float* __restrict__ out)          // [B, A_TOTAL, 5]
{
    // 512 floats of box data (yxyx) + 128 precomputed areas
    __shared__ float smem[NBOX * 4 + NBOX];

    const int b = blockIdx.y;
    const int g = blockIdx.x * BLK + threadIdx.x;   // global anchor index

    // ---- Stage this batch's 128 boxes (2KB) into LDS via the Tensor Data Mover ----
    if (threadIdx.x < 32) {  // wave 0 only (wave32)
        unsigned long long gaddr =
            (unsigned long long)(size_t)(boxes + (size_t)b * NBOX * 4);
        unsigned lds_off = (unsigned)(size_t)&smem[0];  // low 32 bits of flat addr = LDS offset

        // D# group 0: count=1 | lds_addr | global_addr[56:0] | type=2
        tdm_u4 g0;
        g0.x = 1u;
        g0.y = lds_off;
        g0.z = (unsigned)(gaddr & 0xFFFFFFFFull);
        g0.w = (unsigned)((gaddr >> 32) & 0x01FFFFFFull) | (2u << 30);

        // D# group 1: 1-row tensor of 512 x 4-byte elements, tile = whole row
        tdm_i8 g1;
        g1[0] = (2 << 16);        // workgroup_mask=0, data_size=2 (4 bytes)
        g1[1] = (512 << 16);      // tensor_dim0[15:0] = 512 (bits 63:48)
        g1[2] = (1 << 16);        // tensor_dim0[31:16]=0 ; tensor_dim1[15:0] = 1
        g1[3] = (512 << 16);      // tensor_dim1[31:16]=0 ; tile_dim0 = 512
        g1[4] = 1;                // tile_dim1 = 1 ; tile_dim2 = 0
        g1[5] = 512;              // tensor_dim0_stride[31:0] = 512
        g1[6] = 0;
        g1[7] = 0;

        tdm_i4 gz = {0, 0, 0, 0};
#if __clang_major__ >= 23
        tdm_i8 gz8 = {0, 0, 0, 0, 0, 0, 0, 0};
        __builtin_amdgcn_tensor_load_to_lds(g0, g1, gz, gz, gz8, 0);
#else
        __builtin_amdgcn_tensor_load_to_lds(g0, g1, gz, gz, 0);
#endif
        __builtin_amdgcn_s_wait_tensorcnt(0);
    }
    __syncthreads();

    // ---- Precompute per-box areas (area_b) once per block ----
    if (threadIdx.x < NBOX) {
        const float4 bb = ((const float4*)smem)[threadIdx.x];
        smem[NBOX * 4 + threadIdx.x] = (bb.z - bb.x) * (bb.w - bb.y);
    }
    __syncthreads();

    // ---- Anchor geometry from flat index (per-level constant divisors) ----
    float cy, cx, a;
    if (g < 25600) {                       // stride 4, 160x160, a=16
        int i = g;          int r = i / 160; int c = i - r * 160;
        cy = r * 4.0f;  cx = c * 4.0f;  a = 16.0f;
    } else if (g < 32000) {                // stride 8, 80x80, a=32
        int i = g - 25600;  int r = i / 80;  int c = i - r * 80;
        cy = r * 8.0f;  cx = c * 8.0f;  a = 32.0f;
    } else if (g < 33600) {                // stride 16, 40x40, a=64
        int i = g - 32000;  int r = i / 40;  int c = i - r * 40;
        cy = r * 16.0f; cx = c * 16.0f; a = 64.0f;
    } else {                               // stride 32, 20x20, a=128
        int i = g - 33600;  int r = i / 20;  int c = i - r * 20;
        cy = r * 32.0f; cx = c * 32.0f; a = 128.0f;
    }
    const float ha = 0.5f * a;
    const float ay1 = cy - ha, ax1 = cx - ha, ay2 = cy + ha, ax2 = cx + ha;
    const float area_a = a * a;

    // ---- IoU max/argmax over 128 boxes, division-free via cross-multiplication ----
    const float4* __restrict__ sb = (const float4*)smem;
    const float*  __restrict__ sarea = smem + NBOX * 4;

    float best_inter = -1.0f;   // guarantees first iteration wins
    float best_denom = 1.0f;
    int   bestn = 0;

#pragma unroll 4
    for (int n = 0; n < NBOX; ++n) {
        const float4 bb = sb[n];            // wave-uniform LDS broadcast
        const float ymin = fmaxf(ay1, bb.x);
        const float xmin = fmaxf(ax1, bb.y);
        const float ymax = fminf(ay2, bb.z);
        const float xmax = fminf(ax2, bb.w);
        const float ih = fmaxf(ymax - ymin, 0.0f);
        const float iw = fmaxf(xmax - xmin, 0.0f);
        const float inter = ih * iw;
        const float denom = area_a + sarea[n] - inter;   // union area, > 0
        // iou_n > iou_best  <=>  inter*best_denom > best_inter*denom  (denoms > 0)
        if (inter * best_denom > best_inter * denom) {
            best_inter = inter;
            best_denom = denom;
            bestn = n;
        }
    }

    const float best_iou = best_inter / best_denom;      // single divide per anchor

    const float4 mb = sb[bestn];                         // matched box (yxyx)
    const float mcy = 0.5f * (mb.x + mb.z);
    const float mcx = 0.5f * (mb.y + mb.w);
    const float mh  = mb.z - mb.x;
    const float mw  = mb.w - mb.y;
    const float inv_a = 1.0f / a;                        // exact (power of two)

    if (g < A_TOTAL) {
        float* o = out + ((size_t)b * A_TOTAL + g) * 5;
        o[0] = best_iou;
        o[1] = (mcy - cy) * inv_a;
        o[2] = (mcx - cx) * inv_a;
        o[3] = (mh - a) * inv_a;
        o[4] = (mw - a) * inv_a;
    }
}

extern "C" void kernel_launch(void* const* d_in, const int* in_sizes, int n_in,
                              void* d_out, int out_size, void* d_ws, size_t ws_size,
                              hipStream_t stream) {
    const float* boxes = (const float*)d_in[0];   // [16, 128, 4] f32
    float* out = (float*)d_out;                   // [16, 34000, 5] f32
    (void)in_sizes; (void)n_in; (void)out_size; (void)d_ws; (void)ws_size;

    dim3 grid((A_TOTAL + BLK - 1) / BLK, 16, 1);  // (133, 16)
    MultiAnchor_encode_kernel<<<grid, BLK, 0, stream>>>(boxes, out);
}